// QuanvolutionFilter_65481071401511
// MI455X (gfx1250) — compile-verified
//
#include <hip/hip_runtime.h>
#include <hip/hip_bf16.h>

typedef float v2f __attribute__((ext_vector_type(2)));
typedef float v8f __attribute__((ext_vector_type(8)));

#define NQ  4
#define DIM 16

// ---------------------------------------------------------------------------
// Kernel 1 (one-shot setup, deliberately rolled/compact): build the four real
// 16x16 matrices A_w from the weights.
//   ev_w = r^T A_w r  with  A_w[i][j] = Re( (U^H D_w U)[i][j] * i^(popc(i)-popc(j)) )
// Lanes 0..15 simulate the fixed circuit on basis state |i> (column i of U),
// then all 256 threads assemble A.
// ---------------------------------------------------------------------------
__global__ void __launch_bounds__(256) build_A_kernel(const float* __restrict__ W,
                                                      float* __restrict__ A) {
    __shared__ float Ur[DIM][DIM];
    __shared__ float Ui[DIM][DIM];
    const int tid = threadIdx.x;

    if (tid < DIM) {
        float sre[DIM], sim[DIM];
        #pragma clang loop unroll(disable)
        for (int p = 0; p < DIM; ++p) { sre[p] = (p == tid) ? 1.0f : 0.0f; sim[p] = 0.0f; }

        #pragma clang loop unroll(disable)
        for (int g = 0; g < 4; ++g) {              // g = template*2 + layer
            const int ll = g & 1;
            // Rot(phi, theta, omega) on each wire
            #pragma clang loop unroll(disable)
            for (int wq = 0; wq < NQ; ++wq) {
                const int base = (g * NQ + wq) * 3;
                const float phi = W[base + 0], th = W[base + 1], om = W[base + 2];
                const float ct = __cosf(th * 0.5f), sv = __sinf(th * 0.5f);
                const float pp = (phi + om) * 0.5f, mm = (phi - om) * 0.5f;
                const float cp = __cosf(pp), sp = __sinf(pp);
                const float cm = __cosf(mm), sm = __sinf(mm);
                // u00 = e^{-ip}ct, u01 = -e^{+im}st, u10 = e^{-im}st, u11 = e^{+ip}ct
                const float u00r =  cp * ct, u00i = -sp * ct;
                const float u01r = -cm * sv, u01i = -sm * sv;
                const float u10r =  cm * sv, u10i = -sm * sv;
                const float u11r =  cp * ct, u11i =  sp * ct;
                const int mask = 8 >> wq;          // wire wq <-> bit (3-wq)
                #pragma clang loop unroll(disable)
                for (int p = 0; p < DIM; ++p) {
                    if (p & mask) continue;
                    const int q = p | mask;
                    const float ar = sre[p], ai = sim[p];
                    const float br = sre[q], bi = sim[q];
                    sre[p] = u00r*ar - u00i*ai + u01r*br - u01i*bi;
                    sim[p] = u00r*ai + u00i*ar + u01r*bi + u01i*br;
                    sre[q] = u10r*ar - u10i*ai + u11r*br - u11i*bi;
                    sim[q] = u10r*ai + u10i*ar + u11r*bi + u11i*br;
                }
            }
            // ring of CNOTs, range r = ll % (NQ-1) + 1  (= ll+1 for ll in {0,1})
            const int rr = ll + 1;
            #pragma clang loop unroll(disable)
            for (int wq = 0; wq < NQ; ++wq) {
                const int cmsk = 8 >> wq, tmsk = 8 >> ((wq + rr) & 3);
                #pragma clang loop unroll(disable)
                for (int p = 0; p < DIM; ++p) {
                    if ((p & cmsk) && !(p & tmsk)) {
                        const int q = p | tmsk;
                        float tr = sre[p], ti = sim[p];
                        sre[p] = sre[q]; sim[p] = sim[q];
                        sre[q] = tr;     sim[q] = ti;
                    }
                }
            }
        }
        #pragma clang loop unroll(disable)
        for (int p = 0; p < DIM; ++p) { Ur[p][tid] = sre[p]; Ui[p][tid] = sim[p]; }
    }
    __syncthreads();

    const int i = tid >> 4, j = tid & 15;
    const int ph = (__popc(i) - __popc(j)) & 3;
    #pragma clang loop unroll(disable)
    for (int w = 0; w < NQ; ++w) {
        float re = 0.0f, im = 0.0f;
        #pragma clang loop unroll(disable)
        for (int p = 0; p < DIM; ++p) {
            const float sgn = ((p >> (3 - w)) & 1) ? -1.0f : 1.0f;
            const float uri = Ur[p][i], uii = Ui[p][i];
            const float urj = Ur[p][j], uij = Ui[p][j];
            re += sgn * (uri * urj + uii * uij);   // Re(conj(U_pi) U_pj) * sign
            im += sgn * (uri * uij - uii * urj);   // Im(conj(U_pi) U_pj) * sign
        }
        const float val = (ph == 0) ? re : (ph == 1) ? -im : (ph == 2) ? -re : im;
        A[w * 256 + i * 16 + j] = val;
    }
}

// Fixed cross-half swap (lane ^= 16) via ds_swizzle group-of-32 mode:
// offset = {0, xor=0x10, or=0x00, and=0x1f} -> 0x401f (SWAPX16). One DS op,
// no address VALU (vs generic __shfl_xor lowering).
__device__ __forceinline__ float swap_half(float v) {
    return __int_as_float(__builtin_amdgcn_ds_swizzle(__float_as_int(v), 0x401f));
}

// ---------------------------------------------------------------------------
// Kernel 2: main quanvolution. Each wave handles 32 patches (2 WMMA tiles).
//   Yt_w = A_w x R^T  via V_WMMA_F32_16X16X4_F32 (patches on N!),
// so lane (n,h) holds Yt[v+8h][n] -> ev[n] = sum_i r_n[i]*Yt[i][n] is a
// lane-local 8-FMA dot per half + ONE swizzle(^16) to combine halves.
// Output store is fully coalesced float2 per lane, no predication.
// ---------------------------------------------------------------------------
__global__ void __launch_bounds__(256) quanv_kernel(const float* __restrict__ x,
                                                    const float* __restrict__ A,
                                                    float* __restrict__ out) {
    __shared__ float rlds[8][32][18];   // stride 18 -> conflict-free rows

    const int lane = threadIdx.x & 31;
    const int wv   = threadIdx.x >> 5;
    const int h = lane >> 4;            // lane half (K offset select)
    const int n = lane & 15;            // row (A) / column-patch (B,D) index

    // Loop-invariant A-operand fragments: A_w as M=16 x K=16.
    // f32 A 16x4 layout: lane holds row i = n; VGPR0/1 = cols 4c+2h, 4c+2h+1.
    v2f Af[4][4];
    #pragma unroll
    for (int w = 0; w < 4; ++w)
        #pragma unroll
        for (int c = 0; c < 4; ++c) {
            const float* ap = A + w * 256 + n * 16 + 4 * c + 2 * h;
            Af[w][c][0] = ap[0];
            Af[w][c][1] = ap[1];
        }

    // One patch per lane: 4096 imgs * 196 patches = 802816 = gridDim*256 exactly.
    const int patch = blockIdx.x * 256 + threadIdx.x;
    const int b  = patch / 196;
    const int pr = patch - b * 196;
    const int pi = pr / 14;
    const int pj = pr - pi * 14;
    const float* px = x + b * 784 + pi * 56 + pj * 2;
    const float a0 = px[0], a1 = px[1], a2 = px[28], a3 = px[29];

    const float c0 = __cosf(a0 * 0.5f), s0 = __sinf(a0 * 0.5f);
    const float c1 = __cosf(a1 * 0.5f), s1 = __sinf(a1 * 0.5f);
    const float c2 = __cosf(a2 * 0.5f), s2 = __sinf(a2 * 0.5f);
    const float c3 = __cosf(a3 * 0.5f), s3 = __sinf(a3 * 0.5f);
    // r_i = hi[i>>2] * lo[i&3]   (wire w <-> bit 3-w)
    const float hi0 = c0*c1, hi1 = c0*s1, hi2 = s0*c1, hi3 = s0*s1;
    const float lo0 = c2*c3, lo1 = c2*s3, lo2 = s2*c3, lo3 = s2*s3;
    {
        float* rp = &rlds[wv][lane][0];
        rp[ 0]=hi0*lo0; rp[ 1]=hi0*lo1; rp[ 2]=hi0*lo2; rp[ 3]=hi0*lo3;
        rp[ 4]=hi1*lo0; rp[ 5]=hi1*lo1; rp[ 6]=hi1*lo2; rp[ 7]=hi1*lo3;
        rp[ 8]=hi2*lo0; rp[ 9]=hi2*lo1; rp[10]=hi2*lo2; rp[11]=hi2*lo3;
        rp[12]=hi3*lo0; rp[13]=hi3*lo1; rp[14]=hi3*lo2; rp[15]=hi3*lo3;
    }
    __syncthreads();

    #pragma unroll
    for (int t = 0; t < 2; ++t) {
        const float* myr = &rlds[wv][t * 16 + n][0];   // r of patch (tile,n)

        // B-operand fragments: R^T as K=16 x N=16, column p = n.
        // f32 B 4x16 layout: VGPR0/1 = rows 4c+2h, 4c+2h+1 at column n.
        v2f Bf[4];
        #pragma unroll
        for (int c = 0; c < 4; ++c) {
            Bf[c][0] = myr[4 * c + 2 * h];
            Bf[c][1] = myr[4 * c + 2 * h + 1];
        }

        v8f acc[4] = { {0}, {0}, {0}, {0} };
        #pragma unroll
        for (int c = 0; c < 4; ++c) {
            #pragma unroll
            for (int w = 0; w < 4; ++w) {
                acc[w] = __builtin_amdgcn_wmma_f32_16x16x4_f32(
                    false, Af[w][c], false, Bf[c], (short)0, acc[w], false, false);
            }
        }

        // lane-local half-dot: rows 8h..8h+7 of Yt against r_n
        float rv[8];
        #pragma unroll
        for (int v = 0; v < 8; ++v) rv[v] = myr[8 * h + v];

        float ev[4];
        #pragma unroll
        for (int w = 0; w < 4; ++w) {
            float d = 0.0f;
            #pragma unroll
            for (int v = 0; v < 8; ++v) d = fmaf(rv[v], acc[w][v], d);
            d += swap_half(d);             // combine the two halves
            ev[w] = d;
        }

        // out flat index = patch*4 + w ; tile of 16 patches -> 64 contiguous
        // floats. Lane (n,h) writes {ev[2h], ev[2h+1]} at n*4 + 2h: 32 lanes
        // x 8B = whole 256B tile, coalesced, no predication.
        const float o0 = h ? ev[2] : ev[0];
        const float o1 = h ? ev[3] : ev[1];
        float* outT = out + blockIdx.x * 1024 + wv * 128 + t * 64;
        *(float2*)(outT + n * 4 + 2 * h) = make_float2(o0, o1);
    }
}

// ---------------------------------------------------------------------------
extern "C" void kernel_launch(void* const* d_in, const int* in_sizes, int n_in,
                              void* d_out, int out_size, void* d_ws, size_t ws_size,
                              hipStream_t stream) {
    const float* x = (const float*)d_in[0];   // (4096, 784) f32
    const float* W = (const float*)d_in[1];   // (2,2,4,3)   f32
    float* out = (float*)d_out;               // (4096, 784) f32
    float* A   = (float*)d_ws;                // 4 * 16 * 16 f32 = 4 KB scratch

    build_A_kernel<<<1, 256, 0, stream>>>(W, A);
    // 4096*196 patches / 256 per block = 3136 blocks
    quanv_kernel<<<3136, 256, 0, stream>>>(x, A, out);
}